// MultiSplineBinaryEncoding_84645215469599
// MI455X (gfx1250) — compile-verified
//
#include <hip/hip_runtime.h>
#include <cstdint>

#define NEMB 100
#define TBINS 3746
#define FPB 8
#define ROWS 24                   // 12 (b,c) terms * 2 rows (w0,w1)
#define BUF_FLOATS (ROWS * NEMB)  // 2400 floats = 9600 B per fragment buffer

// ---- CDNA5 async global->LDS copy (per-lane 16B), tracked by ASYNCcnt ----
__device__ __forceinline__ void async_copy_b128(uint32_t lds_byte, uint64_t gaddr) {
  // VDST = LDS byte offset (low 32 bits of flat shared pointer == AS(3) offset),
  // VADDR = 64-bit global address, GV mode (no SADDR).
  asm volatile("global_load_async_to_lds_b128 %0, %1, off"
               :: "v"(lds_byte), "v"(gaddr) : "memory");
}

__device__ __forceinline__ void wait_async_le5() {
  asm volatile("s_wait_asynccnt 0x5" ::: "memory");
}
__device__ __forceinline__ void wait_async_le0() {
  asm volatile("s_wait_asynccnt 0x0" ::: "memory");
}

// Exact u / bw for u in [0, 199999], bw in {100,200,500,1000,2000,5000}
// via q = (u * ceil(2^37/bw)) >> 37  (error bound: bw <= 5000 << 2^37/200000)
__device__ __forceinline__ unsigned udiv_bw(unsigned u, int b) {
  const unsigned M =
      b == 0 ? 1374389535u :   // 100
      b == 1 ? 687194768u  :   // 200
      b == 2 ? 274877907u  :   // 500
      b == 3 ? 137438954u  :   // 1000
      b == 4 ? 68719477u   :   // 2000
               27487791u;      // 5000
  return (unsigned)(((unsigned long long)u * M) >> 37);
}

__device__ __forceinline__ int cum_of(int b) {
  // nbins = [2001,1001,401,201,101,41] -> exclusive prefix sums
  return b == 0 ? 0 : b == 1 ? 2001 : b == 2 ? 3002 :
         b == 3 ? 3403 : b == 4 ? 3604 : 3705;
}

struct FragInfo { int u0, u1, rid; };

__device__ __forceinline__ FragInfo load_info(int f,
                                              const int* __restrict__ coords,
                                              const int* __restrict__ indptr,
                                              const int* __restrict__ regions,
                                              int R) {
  FragInfo fi;
  int c0 = coords[2 * f + 0];
  int c1 = coords[2 * f + 1];
  c0 = min(max(c0, -100000), 99999);
  c1 = min(max(c1, -100000), 99999);
  fi.u0 = c0 + 100000;  // in [0, 199999]
  fi.u1 = c1 + 100000;
  // seg = last s in [0,R-1] with indptr[s] <= f  (uniform per block -> scalar-ish)
  int lo = 0, hi = R;
  while (hi - lo > 1) {
    int mid = (lo + hi) >> 1;
    if (indptr[mid] <= f) lo = mid; else hi = mid;
  }
  fi.rid = regions[lo];
  return fi;
}

// Stage one fragment's 24 rows (24 x 100 f32) into LDS via async B128 copies.
// Wave w owns row slots [6w, 6w+6) = 150 chunks -> exactly 5 async issues/wave.
__device__ __forceinline__ void stage_frag(const FragInfo& fi,
                                           const float* __restrict__ w,
                                           float* buf, int wave, int lane) {
  uint64_t wbase = (uint64_t)(uintptr_t)w +
                   (uint64_t)fi.rid * ((uint64_t)TBINS * NEMB * 4u);
  uint32_t lbase = (uint32_t)(uintptr_t)buf;  // LDS byte offset
  #pragma unroll
  for (int j = 0; j < 5; ++j) {
    int m  = j * 32 + lane;        // 0..159 within this wave's work
    int q  = m / 25;               // row within wave's 6 slots
    int ch = m - q * 25;           // 16B chunk within row (25 per row)
    int k  = wave * 6 + q;         // global row slot 0..23
    int t  = k >> 1;               // term 0..11
    int s  = k & 1;                // 0 -> w0 row, 1 -> w1 row
    int b  = t >> 1;               // binwidth 0..5
    unsigned u = (t & 1) ? (unsigned)fi.u1 : (unsigned)fi.u0;
    int row = (int)udiv_bw(u, b) + cum_of(b) + s;       // bin row, row+1 in range
    uint64_t ga = wbase + (uint64_t)((unsigned)row * (NEMB * 4u)) + (unsigned)ch * 16u;
    uint32_t la = lbase + (unsigned)k * (NEMB * 4u) + (unsigned)ch * 16u;
    if (m < 150) async_copy_b128(la, ga);   // every wave: 5 issues, >=1 active lane
  }
}

__device__ __forceinline__ void compute_frag(const FragInfo& fi,
                                             const float* buf,
                                             float* __restrict__ outrow,
                                             int tid) {
  if (tid < NEMB) {
    float acc = 0.0f;
    #pragma unroll
    for (int b = 0; b < 6; ++b) {
      const int bw = (b == 0) ? 100 : (b == 1) ? 200 : (b == 2) ? 500 :
                     (b == 3) ? 1000 : (b == 4) ? 2000 : 5000;
      #pragma unroll
      for (int c = 0; c < 2; ++c) {
        unsigned u = c ? (unsigned)fi.u1 : (unsigned)fi.u0;
        float alpha = (float)(u % (unsigned)bw) / (float)bw;
        int t = b * 2 + c;
        float w0 = buf[(2 * t + 0) * NEMB + tid];
        float w1 = buf[(2 * t + 1) * NEMB + tid];
        acc += w0 * (1.0f - alpha) + w1 * alpha;
      }
    }
    outrow[tid] = acc;
  }
}

extern "C" __global__ void __launch_bounds__(128)
mse_gather_kernel(const int* __restrict__ coordinates,
                  const int* __restrict__ indptr,
                  const int* __restrict__ regions_oi,
                  const float* __restrict__ w,
                  float* __restrict__ out,
                  int N, int R) {
  __shared__ __align__(16) float smem[2 * BUF_FLOATS];  // 19.2 KB double buffer
  const int tid  = threadIdx.x;
  const int wave = tid >> 5;
  const int lane = tid & 31;
  const int f0   = (int)blockIdx.x * FPB;
  if (f0 >= N) return;

  FragInfo cur = load_info(f0, coordinates, indptr, regions_oi, R);
  stage_frag(cur, w, smem, wave, lane);   // ASYNCcnt: +5 per wave

  #pragma unroll 1
  for (int i = 0; i < FPB; ++i) {
    int f = f0 + i;
    if (f >= N) break;                    // uniform: f depends only on blockIdx,i
    FragInfo nxt = cur;
    bool has_next = (i + 1 < FPB) && (f + 1 < N);  // uniform per block
    if (has_next) {
      nxt = load_info(f + 1, coordinates, indptr, regions_oi, R);
      stage_frag(nxt, w, smem + ((i + 1) & 1) * BUF_FLOATS, wave, lane);  // +5
      wait_async_le5();   // 10 outstanding -> <=5 guarantees frag i's copies done
    } else {
      wait_async_le0();   // drain the final stage
    }
    __syncthreads();      // all waves' copies for frag i are in LDS
    compute_frag(cur, smem + (i & 1) * BUF_FLOATS, out + (size_t)f * NEMB, tid);
    __syncthreads();      // done reading buf[i&1] before it is restaged
    cur = nxt;
  }
}

extern "C" void kernel_launch(void* const* d_in, const int* in_sizes, int n_in,
                              void* d_out, int out_size, void* d_ws, size_t ws_size,
                              hipStream_t stream) {
  const int*   coordinates = (const int*)d_in[0];   // (N,2) int32
  const int*   indptr      = (const int*)d_in[1];   // (R+1,) int32
  const int*   regions_oi  = (const int*)d_in[2];   // (R,)   int32
  const float* w           = (const float*)d_in[3]; // (256, 3746, 100) f32
  float* out = (float*)d_out;                       // (N, 100) f32

  int N = in_sizes[0] / 2;
  int R = in_sizes[1] - 1;
  int grid = (N + FPB - 1) / FPB;
  hipLaunchKernelGGL(mse_gather_kernel, dim3(grid), dim3(128), 0, stream,
                     coordinates, indptr, regions_oi, w, out, N, R);
}